// Model1_31421980737663
// MI455X (gfx1250) — compile-verified
//
#include <hip/hip_runtime.h>
#include <math.h>

typedef float v2f __attribute__((ext_vector_type(2)));
typedef float v4f __attribute__((ext_vector_type(4)));
typedef float v8f __attribute__((ext_vector_type(8)));

// ---------------------------------------------------------------------------
// Degree / normalization
// ---------------------------------------------------------------------------
__global__ void zero_f32_kernel(float* __restrict__ p, int n) {
    int i = blockIdx.x * blockDim.x + threadIdx.x;
    if (i < n) p[i] = 0.0f;
}

__global__ void degree_kernel(const int* __restrict__ dst, int E, float* __restrict__ deg) {
    int e = blockIdx.x * blockDim.x + threadIdx.x;
    if (e < E) atomicAdd(&deg[dst[e]], 1.0f);
}

__global__ void dinv_kernel(float* __restrict__ deg, int n) {
    int i = blockIdx.x * blockDim.x + threadIdx.x;
    if (i < n) deg[i] = rsqrtf(deg[i] + 1.0f);  // +1 for self loop; always > 0
}

// ---------------------------------------------------------------------------
// WMMA fp32 GEMM: H[M x OUT_C] = X[M x IN_C] @ W[IN_C x OUT_C] (+ bias)
// One wave computes a 16-row strip across all column tiles.
// V_WMMA_F32_16X16X4_F32 layouts (ISA 7.12.2):
//   A 16x4 : lane&15 = M row; VGPR0 holds K = 2*(lane>>4), VGPR1 holds K+1
//   B 4x16 : lane&15 = N col; VGPR0 holds K = 2*(lane>>4), VGPR1 holds K+1
//   C/D    : lane&15 = N col; VGPR v holds M = v + 8*(lane>>4)
// ---------------------------------------------------------------------------
template <int IN_C, int OUT_C, bool BIAS>
__global__ void gemm_wmma_kernel(const float* __restrict__ X,
                                 const float* __restrict__ W,
                                 const float* __restrict__ bias,
                                 float* __restrict__ H, int M) {
    constexpr int NCT = (OUT_C + 15) / 16;  // column tiles
    constexpr int KT  = (IN_C + 3) / 4;     // K steps of 4

    int wave = (blockIdx.x * blockDim.x + threadIdx.x) >> 5;
    int lane = threadIdx.x & 31;
    int numRowTiles = (M + 15) >> 4;
    if (wave >= numRowTiles) return;  // wave-uniform: EXEC all-ones inside

    const int row0  = wave << 4;
    const int aRow  = row0 + (lane & 15);
    const int kHalf = (lane >> 4) << 1;  // 0 or 2
    const int col   = lane & 15;

    v8f acc[NCT] = {};

    const float* __restrict__ xrow = X + (size_t)aRow * IN_C;
    for (int kt = 0; kt < KT; ++kt) {
        const int k = kt * 4 + kHalf;
        v2f a;
        a.x = (k     < IN_C) ? xrow[k]     : 0.0f;
        a.y = (k + 1 < IN_C) ? xrow[k + 1] : 0.0f;
#pragma unroll
        for (int t = 0; t < NCT; ++t) {
            const int c = t * 16 + col;
            const bool cok = (c < OUT_C);
            v2f b;
            b.x = (cok && (k     < IN_C)) ? W[(size_t)k       * OUT_C + c] : 0.0f;
            b.y = (cok && (k + 1 < IN_C)) ? W[(size_t)(k + 1) * OUT_C + c] : 0.0f;
            acc[t] = __builtin_amdgcn_wmma_f32_16x16x4_f32(
                false, a, false, b, (short)0, acc[t], false, false);
        }
    }

    const int rowOff = (lane >> 4) << 3;  // 0 or 8
#pragma unroll
    for (int t = 0; t < NCT; ++t) {
        const int c = t * 16 + col;
        if (c < OUT_C) {
            if constexpr (BIAS) {
                const float bv = bias[c];
#pragma unroll
                for (int v = 0; v < 8; ++v) {
                    const int r = row0 + rowOff + v;
                    H[(size_t)r * OUT_C + c] = acc[t][v] + bv;
                }
            } else {
#pragma unroll
                for (int v = 0; v < 8; ++v) {
                    const int r = row0 + rowOff + v;
                    H[(size_t)r * OUT_C + c] = acc[t][v];
                }
            }
        }
    }
}

// ---------------------------------------------------------------------------
// Self-loop + bias init:  out[i,c] = dinv[i]^2 * h[i,c] + b[c]
// ---------------------------------------------------------------------------
template <int C>
__global__ void agg_init_kernel(const float* __restrict__ h,
                                const float* __restrict__ dinv,
                                const float* __restrict__ bias,
                                float* __restrict__ out, int N) {
    long long idx = (long long)blockIdx.x * blockDim.x + threadIdx.x;
    if (idx >= (long long)N * C) return;
    const int i = (int)(idx / C);
    const int c = (int)(idx % C);
    const float d = dinv[i];
    out[idx] = d * d * h[idx] + bias[c];
}

// ---------------------------------------------------------------------------
// Edge scatter: out[dst] += dinv[src]*dinv[dst] * h[src]   (one wave per edge)
// h and out are L2-resident (51 MB << 192 MB), fp32 atomics land in L2.
// For C % 128 == 0 each lane owns 4 contiguous channels -> global_load_b128.
// ---------------------------------------------------------------------------
template <int C>
__global__ void agg_edge_kernel(const float* __restrict__ h,
                                const float* __restrict__ dinv,
                                const int* __restrict__ ei, int E,
                                float* __restrict__ out) {
    int wave = (blockIdx.x * blockDim.x + threadIdx.x) >> 5;
    int lane = threadIdx.x & 31;
    if (wave >= E) return;
    const int s = ei[wave];
    const int d = ei[E + wave];
    const float w = dinv[s] * dinv[d];
    const float* __restrict__ hs = h + (size_t)s * C;
    float* __restrict__ od = out + (size_t)d * C;
    if constexpr ((C & 127) == 0) {
#pragma unroll
        for (int g = 0; g < C / 128; ++g) {
            const int c0 = (g * 32 + lane) * 4;
            const v4f hv = *(const v4f*)(hs + c0);   // global_load_b128, coalesced
            atomicAdd(&od[c0 + 0], w * hv.x);
            atomicAdd(&od[c0 + 1], w * hv.y);
            atomicAdd(&od[c0 + 2], w * hv.z);
            atomicAdd(&od[c0 + 3], w * hv.w);
        }
    } else {
#pragma unroll
        for (int c = lane; c < C; c += 32) {
            atomicAdd(&od[c], w * hs[c]);
        }
    }
}

// ---------------------------------------------------------------------------
// In-place log_softmax over 40 columns, one wave per row (wave32 shuffles)
// ---------------------------------------------------------------------------
__global__ void log_softmax40_kernel(float* __restrict__ out, int N) {
    int wave = (blockIdx.x * blockDim.x + threadIdx.x) >> 5;
    int lane = threadIdx.x & 31;
    if (wave >= N) return;
    float* __restrict__ row = out + (size_t)wave * 40;
    const float v0 = row[lane];                                      // lane < 32 < 40
    const float v1 = (lane + 32 < 40) ? row[lane + 32] : -INFINITY;  // lanes 0..7
    float m = fmaxf(v0, v1);
#pragma unroll
    for (int o = 16; o > 0; o >>= 1) m = fmaxf(m, __shfl_xor(m, o, 32));
    float s = expf(v0 - m) + ((lane + 32 < 40) ? expf(v1 - m) : 0.0f);
#pragma unroll
    for (int o = 16; o > 0; o >>= 1) s += __shfl_xor(s, o, 32);
    const float lse = logf(s) + m;
    row[lane] = v0 - lse;
    if (lane + 32 < 40) row[lane + 32] = v1 - lse;
}

// ---------------------------------------------------------------------------
// Launcher
// ---------------------------------------------------------------------------
extern "C" void kernel_launch(void* const* d_in, const int* in_sizes, int n_in,
                              void* d_out, int out_size, void* d_ws, size_t ws_size,
                              hipStream_t stream) {
    (void)n_in; (void)out_size; (void)ws_size;

    const float* x  = (const float*)d_in[0];
    const float* W1 = (const float*)d_in[1];
    const float* b1 = (const float*)d_in[2];
    const float* W2 = (const float*)d_in[3];
    const float* b2 = (const float*)d_in[4];
    const float* W3 = (const float*)d_in[5];
    const float* b3 = (const float*)d_in[6];
    const float* Wf = (const float*)d_in[7];
    const float* bf = (const float*)d_in[8];
    const int*   ei = (const int*)d_in[9];
    float* out = (float*)d_out;

    const int N = in_sizes[0] / 128;   // 100000
    const int E = in_sizes[9] / 2;     // 1600000

    // Workspace layout (floats): dinv[N] | bufA[N*128] | bufB[N*128]
    float* ws = (float*)d_ws;
    const size_t dinvPad = (((size_t)N + 63) / 64) * 64;
    float* dinv = ws;
    float* bufA = ws + dinvPad;
    float* bufB = bufA + (size_t)N * 128;

    const int TB = 256;
    auto cdiv = [](long long a, long long b) { return (int)((a + b - 1) / b); };

    const int gemmBlocks  = cdiv(((long long)((N + 15) / 16)) * 32, TB);
    const int edgeBlocks  = cdiv((long long)E * 32, TB);
    const int rowBlocks   = cdiv((long long)N * 32, TB);

    // degree -> dinv
    zero_f32_kernel<<<cdiv(N, TB), TB, 0, stream>>>(dinv, N);
    degree_kernel<<<cdiv(E, TB), TB, 0, stream>>>(ei + E, E, dinv);
    dinv_kernel<<<cdiv(N, TB), TB, 0, stream>>>(dinv, N);

    // ---- GCN layer 1: x[N,128] -> bufB[N,128]
    gemm_wmma_kernel<128, 128, false><<<gemmBlocks, TB, 0, stream>>>(x, W1, nullptr, bufA, N);
    agg_init_kernel<128><<<cdiv((long long)N * 128, TB), TB, 0, stream>>>(bufA, dinv, b1, bufB, N);
    agg_edge_kernel<128><<<edgeBlocks, TB, 0, stream>>>(bufA, dinv, ei, E, bufB);

    // ---- GCN layer 2: bufB[N,128] -> bufB[N,128] (via bufA)
    gemm_wmma_kernel<128, 128, false><<<gemmBlocks, TB, 0, stream>>>(bufB, W2, nullptr, bufA, N);
    agg_init_kernel<128><<<cdiv((long long)N * 128, TB), TB, 0, stream>>>(bufA, dinv, b2, bufB, N);
    agg_edge_kernel<128><<<edgeBlocks, TB, 0, stream>>>(bufA, dinv, ei, E, bufB);

    // ---- GCN layer 3: bufB[N,128] -> bufB[N,30] (via bufA)
    gemm_wmma_kernel<128, 30, false><<<gemmBlocks, TB, 0, stream>>>(bufB, W3, nullptr, bufA, N);
    agg_init_kernel<30><<<cdiv((long long)N * 30, TB), TB, 0, stream>>>(bufA, dinv, b3, bufB, N);
    agg_edge_kernel<30><<<edgeBlocks, TB, 0, stream>>>(bufA, dinv, ei, E, bufB);

    // ---- Final linear: bufB[N,30] @ Wf[30,40] + bf -> logits in d_out
    gemm_wmma_kernel<30, 40, true><<<gemmBlocks, TB, 0, stream>>>(bufB, Wf, bf, out, N);

    // ---- log_softmax over 40 columns, in place
    log_softmax40_kernel<<<rowBlocks, TB, 0, stream>>>(out, N);
}